// DarcySolver_62371515072895
// MI455X (gfx1250) — compile-verified
//
#include <hip/hip_runtime.h>
#include <hip/hip_bf16.h>

typedef __attribute__((ext_vector_type(2))) float v2f;
typedef __attribute__((ext_vector_type(8))) float v8f;

namespace {
constexpr int kN     = 512;   // grid nodes
constexpr int kBatch = 128;   // RHS count
constexpr int kLdsStride = 516; // 512 + 4 pad: 516 mod 64 == 4 -> conflict-free column reads
}

// Out[128,512] = Forcing[128,512] x M^T, where M is the analytic inverse of the
// FEM tridiagonal operator (h/2 load scaling folded in), boundaries patched.
// Key identity (exact in fp32, all integer products < 2^24):
//   M[i,j] = scale * min(i,j)*(511-max(i,j)) = scale * (511*min(i,j) - i*j)
// j-boundaries (j=0, j=511) vanish automatically; i-boundary folded into a
// lane-constant scale. B-gen is done with packed-f32 VALU (pk_mul/pk_fma/pk_add)
// so each V_WMMA_F32_16X16X4_F32 costs only ~6 VALU ops of matrix generation.
__global__ __launch_bounds__(256) void darcy_wmma_f32(const float* __restrict__ forcing,
                                                      float* __restrict__ out) {
  __shared__ float sfor[16 * kLdsStride];

  const int bt   = blockIdx.x;     // batch tile 0..7 (16 RHS rows each)
  const int tid  = threadIdx.x;    // 0..255 (8 wave32s)
  const int wave = tid >> 5;       // 0..7
  const int lane = tid & 31;

  // ---- Stage this block's 16x512 forcing slab into LDS (coalesced float4) ----
  {
    const float4* src = (const float4*)(forcing + (size_t)bt * 16 * kN);
    #pragma unroll
    for (int i = 0; i < 8; ++i) {
      int q   = tid + 256 * i;     // float4 index within slab (2048 total)
      int row = q >> 7;            // 128 float4 per 512-float row
      int c4  = q & 127;
      float4 vv = src[q];
      *(float4*)&sfor[row * kLdsStride + c4 * 4] = vv;
    }
  }
  __syncthreads();

  // ISA 16x4 f32 A layout: lane holds row m = lane%16; lanes 0-15 carry K={0,1},
  // lanes 16-31 carry K={2,3} in VGPRs {0,1}.
  const int m  = lane & 15;
  const int kh = (lane >> 4) * 2;

  const float scale  = 5.0f / (511.0f * 511.0f * 511.0f);
  const float sin_pi = sinf(3.14159274101257324f); // sin of fp32 pi, matches reference BC

  for (int tt = 0; tt < 4; ++tt) {
    const int ct      = wave * 4 + tt;  // column tile 0..31
    const int colbase = ct * 16;
    const int icol    = colbase + m;    // this lane's B-matrix output-node index n
    const float icolf = (float)icol;
    // Dirichlet rows 0 and 511 of M are identically zero -> zero the lane scale.
    const float lscale = (icol >= 1 && icol <= 510) ? scale : 0.0f;

    const v2f icv  = {icolf, icolf};
    const v2f lsv  = {lscale, lscale};
    const v2f c511 = {511.0f, 511.0f};
    v2f jv = {(float)kh, (float)(kh + 1)};  // this lane's two K indices (float, exact)

    v8f acc = {};
    for (int k0 = 0; k0 < kN; k0 += 4) {
      // A: forcing tile, 2 fp32 per lane from LDS (8B aligned, bank-conflict free)
      v2f a = *(const v2f*)&sfor[m * kLdsStride + k0 + kh];

      // B[k][n] = lscale * (511*min(icol,j) - icol*j); packed-f32 VALU, exact ints.
      v2f mn;
      mn[0] = fminf(icolf, jv[0]);
      mn[1] = fminf(icolf, jv[1]);
      v2f p = icv * jv;                               // v_pk_mul_f32 (exact)
      v2f q = __builtin_elementwise_fma(c511, mn, -p); // v_pk_fma_f32 (exact)
      v2f b = lsv * q;                                 // v_pk_mul_f32 (one rounding)
      jv += 4.0f;                                      // v_pk_add_f32

      // D = A*B + C, full fp32 matrix op (the only fp32 WMMA shape on CDNA5)
      acc = __builtin_amdgcn_wmma_f32_16x16x4_f32(
          /*neg_a=*/false, a, /*neg_b=*/false, b,
          /*c_mod=*/(short)0, acc, /*reuse_a=*/false, /*reuse_b=*/false);
    }

    // D layout: VGPR v, lane l -> row (v + 8*(l>=16)), col (l&15). Coalesced stores.
    const int nn   = lane & 15;
    const int half = lane >> 4;
    #pragma unroll
    for (int v = 0; v < 8; ++v) {
      const int row = bt * 16 + v + 8 * half;
      const int col = colbase + nn;
      out[(size_t)row * kN + col] = (col == kN - 1) ? sin_pi : acc[v];
    }
  }
}

extern "C" void kernel_launch(void* const* d_in, const int* in_sizes, int n_in,
                              void* d_out, int out_size, void* d_ws, size_t ws_size,
                              hipStream_t stream) {
  (void)in_sizes; (void)n_in; (void)out_size; (void)d_ws; (void)ws_size;
  const float* forcing = (const float*)d_in[0];  // [128, 512] fp32
  float* out = (float*)d_out;                    // [128, 512] fp32
  // 8 blocks (one per batch tile) x 256 threads (8 wave32s); each wave covers
  // four 16x16 column tiles -> 256 tiles total = full 128x512 output.
  darcy_wmma_f32<<<dim3(kBatch / 16), dim3(256), 0, stream>>>(forcing, out);
}